// sum_plus_max_75033078661468
// MI455X (gfx1250) — compile-verified
//
#include <hip/hip_runtime.h>

typedef __attribute__((ext_vector_type(2))) float v2f;
typedef __attribute__((ext_vector_type(8))) float v8f;

#define BN_EPS 1e-3f

// ---------------------------------------------------------------------------
// Workspace layout (floats):
//   [0, S*128)                : segbuf, interleaved per segment row of 128:
//                               even col 2c = running MAX, odd col 2c+1 = SUM
//                               (this interleaving IS the groups=2 shuffle)
//   base = S*128:
//   [base, base+Cin*64)       : W1f  = W1 with BN1 scale folded in
//   [base+640, +64)           : shift1 (beta1 - mean1*scale1)
//   [base+704, +64)           : scale2
//   [base+768, +64)           : shift2
// ---------------------------------------------------------------------------

// Fold BN params into weights / affine terms. One tiny block.
__global__ void prep_kernel(const float* __restrict__ W1,
                            const float* __restrict__ g1, const float* __restrict__ b1,
                            const float* __restrict__ m1, const float* __restrict__ v1,
                            const float* __restrict__ g2, const float* __restrict__ b2,
                            const float* __restrict__ m2, const float* __restrict__ v2,
                            float* __restrict__ W1f, float* __restrict__ shift1,
                            float* __restrict__ scale2, float* __restrict__ shift2,
                            int Cin) {
  int c = threadIdx.x;
  if (c < 64) {
    float s1 = g1[c] * rsqrtf(v1[c] + BN_EPS);
    shift1[c] = b1[c] - m1[c] * s1;
    float s2 = g2[c] * rsqrtf(v2[c] + BN_EPS);
    scale2[c] = s2;
    shift2[c] = b2[c] - m2[c] * s2;
    for (int i = 0; i < Cin; ++i) W1f[i * 64 + c] = W1[i * 64 + c] * s1;
  }
}

// Zero the segment accumulator buffer (0.0f works for both max-u32 and add-f32).
__global__ void zero_kernel(float4* __restrict__ p, int n4) {
  int i = blockIdx.x * blockDim.x + threadIdx.x;
  if (i < n4) p[i] = make_float4(0.f, 0.f, 0.f, 0.f);
}

// Stage 1 fused: per-point linear(10->64) + BN + ReLU, scatter max/sum into
// interleaved segbuf. Post-ReLU values are >= 0 so u32 atomicMax on the float
// bit pattern is exact, and the 0-init matches the reference's empty-segment
// fixup. Segbuf (51 MB) is L2-resident on MI455X (192 MB L2), so the 128M
// atomics run at L2 atomic-unit rate, not HBM.
__global__ void __launch_bounds__(256) stage1_kernel(
    const float* __restrict__ x, const int* __restrict__ seg,
    const float* __restrict__ W1f_g, const float* __restrict__ shift1_g,
    float* __restrict__ segf, int N, int Cin) {
  __shared__ float lds_w[10 * 64];
  __shared__ float lds_sh[64];
  for (int i = threadIdx.x; i < Cin * 64; i += blockDim.x) lds_w[i] = W1f_g[i];
  if (threadIdx.x < 64) lds_sh[threadIdx.x] = shift1_g[threadIdx.x];
  __syncthreads();

  int p = blockIdx.x * blockDim.x + threadIdx.x;
  if (p >= N) return;

  float xi[10];
  #pragma unroll
  for (int i = 0; i < 10; ++i) xi[i] = (i < Cin) ? x[(size_t)p * Cin + i] : 0.f;

  float* row = segf + (size_t)seg[p] * 128;
  #pragma unroll 4
  for (int c = 0; c < 64; ++c) {
    float acc = lds_sh[c];
    #pragma unroll
    for (int i = 0; i < 10; ++i) acc = fmaf(xi[i], lds_w[i * 64 + c], acc);
    float y = fmaxf(acc, 0.f);
    atomicMax((unsigned int*)(row + 2 * c), __float_as_uint(y));   // segment max
    atomicAdd(row + 2 * c + 1, y);                                 // segment sum
  }
}

// Stage 3: [S,128] @ [128,64] with V_WMMA_F32_16X16X4_F32, BN2+ReLU epilogue.
// 256-thread block = 8 waves; each wave owns one 16-row x 64-col tile
// (4 v8f accumulators). W2 staged once per block in LDS, swizzled into
// row-pair-interleaved layout so every B fragment is ONE aligned ds_load_b64:
//   ldsB[(k>>1)*128 + 2*n + (k&1)] = W2[k*64 + n]
// A-frag (16x4 f32, ISA 7.12.2): lane {half=l>>4, lo=l&15}: a[j]=A[lo][2*half+j]
// B-frag mirrored: b[j]=B[2*half+j][lo+16t]; C/D: c[v]=C[v+8*half][lo].
__global__ void __launch_bounds__(256) stage3_kernel(
    const float* __restrict__ segf, const float* __restrict__ W2,
    const float* __restrict__ scale2_g, const float* __restrict__ shift2_g,
    float* __restrict__ out, int S) {
  __shared__ float ldsB[128 * 64];     // 32 KB, pair-interleaved
  __shared__ float ldsScale[64];
  __shared__ float ldsShift[64];
  for (int i = threadIdx.x; i < 128 * 64; i += 256) {
    const int k = i >> 6;              // W2 row
    const int n = i & 63;              // W2 col
    ldsB[(k >> 1) * 128 + 2 * n + (k & 1)] = W2[i];
  }
  if (threadIdx.x < 64) {
    ldsScale[threadIdx.x] = scale2_g[threadIdx.x];
    ldsShift[threadIdx.x] = shift2_g[threadIdx.x];
  }
  __syncthreads();

  const int wave = threadIdx.x >> 5;
  const int lane = threadIdx.x & 31;
  const int tile = blockIdx.x * 8 + wave;   // 16-row tile index
  const int m0 = tile * 16;
  if (m0 >= S) return;                      // whole-wave uniform exit (after barrier)

  const int half = lane >> 4;
  const int lo = lane & 15;

  // A row for this lane; columns start at 2*half within each K-step of 4.
  const float* arow = segf + (size_t)(m0 + lo) * 128 + 2 * half;

  v8f acc0 = {}, acc1 = {}, acc2 = {}, acc3 = {};

  #pragma unroll 4
  for (int k0 = 0; k0 < 128; k0 += 4) {
    v2f a = *(const v2f*)(arow + k0);       // A[lo][k0+2h], A[lo][k0+2h+1]; 8B aligned
    // pair index for B rows {k0+2h, k0+2h+1}
    const float* bp = ldsB + (k0 / 2 + half) * 128 + 2 * lo;
    v2f b0 = *(const v2f*)(bp + 0);         // n = lo
    v2f b1 = *(const v2f*)(bp + 32);        // n = 16 + lo
    v2f b2 = *(const v2f*)(bp + 64);        // n = 32 + lo
    v2f b3 = *(const v2f*)(bp + 96);        // n = 48 + lo
    acc0 = __builtin_amdgcn_wmma_f32_16x16x4_f32(false, a, false, b0, (short)0, acc0, false, false);
    acc1 = __builtin_amdgcn_wmma_f32_16x16x4_f32(false, a, false, b1, (short)0, acc1, false, false);
    acc2 = __builtin_amdgcn_wmma_f32_16x16x4_f32(false, a, false, b2, (short)0, acc2, false, false);
    acc3 = __builtin_amdgcn_wmma_f32_16x16x4_f32(false, a, false, b3, (short)0, acc3, false, false);
  }

  // Epilogue: BN2 (per-output-channel affine) + ReLU, write [16 x 64] tile.
  v8f accs[4] = {acc0, acc1, acc2, acc3};
  #pragma unroll
  for (int t = 0; t < 4; ++t) {
    const int n = 16 * t + lo;
    const float sc = ldsScale[n];
    const float sh = ldsShift[n];
    #pragma unroll
    for (int v = 0; v < 8; ++v) {
      const int m = m0 + v + 8 * half;                // C/D layout
      out[(size_t)m * 64 + n] = fmaxf(fmaf(accs[t][v], sc, sh), 0.f);
    }
  }
}

extern "C" void kernel_launch(void* const* d_in, const int* in_sizes, int n_in,
                              void* d_out, int out_size, void* d_ws, size_t ws_size,
                              hipStream_t stream) {
  const float* inputs = (const float*)d_in[0];
  const float* W1     = (const float*)d_in[1];
  const float* g1     = (const float*)d_in[2];
  const float* b1     = (const float*)d_in[3];
  const float* m1     = (const float*)d_in[4];
  const float* v1     = (const float*)d_in[5];
  const float* W2     = (const float*)d_in[6];
  const float* g2     = (const float*)d_in[7];
  const float* b2     = (const float*)d_in[8];
  const float* m2     = (const float*)d_in[9];
  const float* v2     = (const float*)d_in[10];
  const int*   seg    = (const int*)d_in[11];
  float* out = (float*)d_out;

  const int Cout = in_sizes[2];            // 64
  const int Cin  = in_sizes[1] / Cout;     // 10
  const int N    = in_sizes[0] / Cin;      // 1,000,000
  const int S    = out_size / Cout;        // 100,000

  float* wsf    = (float*)d_ws;
  float* segf   = wsf;                         // S*128 floats
  float* W1f    = wsf + (size_t)S * 128;
  float* shift1 = W1f + 640;
  float* scale2 = shift1 + 64;
  float* shift2 = scale2 + 64;

  // 1) fold BN params
  hipLaunchKernelGGL(prep_kernel, dim3(1), dim3(64), 0, stream,
                     W1, g1, b1, m1, v1, g2, b2, m2, v2,
                     W1f, shift1, scale2, shift2, Cin);

  // 2) zero segment accumulators
  const int n4 = (S * 128) / 4;
  hipLaunchKernelGGL(zero_kernel, dim3((n4 + 255) / 256), dim3(256), 0, stream,
                     (float4*)segf, n4);

  // 3) fused linear1+BN1+ReLU + segment max/sum scatter (shuffle-interleaved)
  hipLaunchKernelGGL(stage1_kernel, dim3((N + 255) / 256), dim3(256), 0, stream,
                     inputs, seg, W1f, shift1, segf, N, Cin);

  // 4) WMMA GEMM [S,128]x[128,64] + BN2 + ReLU, 8 tiles per 256-thread block
  const int tiles = (S + 15) / 16;
  hipLaunchKernelGGL(stage3_kernel, dim3((tiles + 7) / 8), dim3(256), 0, stream,
                     segf, W2, scale2, shift2, out, S);
}